// PrimitiveProj_75685913690511
// MI455X (gfx1250) — compile-verified
//
#include <hip/hip_runtime.h>
#include <math.h>

typedef __attribute__((ext_vector_type(2))) float v2f;
typedef __attribute__((ext_vector_type(8))) float v8f;

#define C_DIM 256
#define M_DIM 16
#define EPSN 1e-12f

// ---------------------------------------------------------------------------
// Kernel 1: row-wise L2 normalize prototypes and memory_bank.
// One 256-thread block per row; blocks [0,N) -> prototypes, [N,2N) -> memory_bank.
// ---------------------------------------------------------------------------
__global__ __launch_bounds__(256) void normalize_rows(
    const float* __restrict__ proto, const float* __restrict__ mbank,
    float* __restrict__ pn, float* __restrict__ mbn, int N) {
  __shared__ float red[256];
  const int b = blockIdx.x;
  const int t = threadIdx.x;
  const float* src = (b < N) ? proto : mbank;
  float* dst       = (b < N) ? pn    : mbn;
  const int row    = (b < N) ? b     : (b - N);
  float x = src[(size_t)row * C_DIM + t];
  red[t] = x * x;
  __syncthreads();
  for (int s = 128; s > 0; s >>= 1) {
    if (t < s) red[t] += red[t + s];
    __syncthreads();
  }
  float invn = 1.0f / fmaxf(sqrtf(red[0]), EPSN);
  dst[(size_t)row * C_DIM + t] = x * invn;
}

// ---------------------------------------------------------------------------
// Kernel 2: per-class hardest-pair mining. One 256-thread block per class.
// Normalizes 16x256 features in LDS, builds 16x16 Gram, flattened argmin with
// first-occurrence semantics (min_row = pos % M, min_col = pos / M), writes
// Q[2n+0] = f[min_row], Q[2n+1] = f[min_col].
// ---------------------------------------------------------------------------
__global__ __launch_bounds__(256) void mine_hardest(
    const float* __restrict__ F, float* __restrict__ Q) {
  __shared__ float sF[M_DIM * C_DIM];   // normalized features, 16 KB
  __shared__ float sSim[M_DIM * M_DIM];
  __shared__ float sPart[256];
  __shared__ float sInv[M_DIM];
  __shared__ int sRow, sCol;

  const int n = blockIdx.x;
  const int t = threadIdx.x;
  const int r  = t >> 4;          // row 0..15
  const int cg = (t & 15) * 16;   // 16-col chunk
  const float* base = F + (size_t)n * M_DIM * C_DIM;

  float loc[16];
  float ss = 0.0f;
#pragma unroll
  for (int u = 0; u < 16; ++u) {
    float x = base[(size_t)r * C_DIM + cg + u];
    loc[u] = x;
    ss += x * x;
  }
  sPart[t] = ss;
  __syncthreads();
  if (t < M_DIM) {
    float s = 0.0f;
#pragma unroll
    for (int u = 0; u < 16; ++u) s += sPart[t * 16 + u];
    sInv[t] = 1.0f / fmaxf(sqrtf(s), EPSN);
  }
  __syncthreads();
  const float inv = sInv[r];
#pragma unroll
  for (int u = 0; u < 16; ++u) sF[r * C_DIM + cg + u] = loc[u] * inv;
  __syncthreads();

  // 16x16 Gram: thread t computes sim[i][j]
  {
    const int i = t >> 4, j = t & 15;
    float acc = 0.0f;
    for (int c = 0; c < C_DIM; ++c)
      acc += sF[i * C_DIM + c] * sF[j * C_DIM + c];
    sSim[i * M_DIM + j] = acc;
  }
  __syncthreads();
  if (t == 0) {
    float best = sSim[0];
    int bp = 0;
    for (int p = 1; p < M_DIM * M_DIM; ++p)
      if (sSim[p] < best) { best = sSim[p]; bp = p; }   // strict < : first occurrence
    sRow = bp % M_DIM;   // = j
    sCol = bp / M_DIM;   // = i
  }
  __syncthreads();
  float* q0 = Q + (size_t)(2 * n) * C_DIM;
  q0[t]         = sF[sRow * C_DIM + t];
  q0[C_DIM + t] = sF[sCol * C_DIM + t];
}

// ---------------------------------------------------------------------------
// fp32 WMMA 16x16x4 tile engine: one wave computes a 16x64 strip (4 acc tiles),
// K = 256. A-lane layout: M = lane&15, K-pair = k + 2*(lane>=16); B symmetric.
// ---------------------------------------------------------------------------
__device__ __forceinline__ void wmma_strip16x64(
    const float* __restrict__ A, const float* __restrict__ B,
    int mrow, int n0, int lane, v8f& c0, v8f& c1, v8f& c2, v8f& c3) {
  const int koff = (lane >> 4) << 1;  // 0 or 2
  const int nl   = lane & 15;
  const float* ap  = A + (size_t)mrow * C_DIM + koff;
  const float* bp0 = B + (size_t)(n0 + nl) * C_DIM + koff;
  const float* bp1 = bp0 + 16 * C_DIM;
  const float* bp2 = bp0 + 32 * C_DIM;
  const float* bp3 = bp0 + 48 * C_DIM;
#pragma unroll 4
  for (int k = 0; k < C_DIM; k += 4) {
    v2f a  = *(const v2f*)(ap + k);
    v2f b0 = *(const v2f*)(bp0 + k);
    v2f b1 = *(const v2f*)(bp1 + k);
    v2f b2 = *(const v2f*)(bp2 + k);
    v2f b3 = *(const v2f*)(bp3 + k);
    c0 = __builtin_amdgcn_wmma_f32_16x16x4_f32(false, a, false, b0, (short)0, c0, false, false);
    c1 = __builtin_amdgcn_wmma_f32_16x16x4_f32(false, a, false, b1, (short)0, c1, false, false);
    c2 = __builtin_amdgcn_wmma_f32_16x16x4_f32(false, a, false, b2, (short)0, c2, false, false);
    c3 = __builtin_amdgcn_wmma_f32_16x16x4_f32(false, a, false, b3, (short)0, c3, false, false);
  }
}

// ---------------------------------------------------------------------------
// Kernel 3: sim_prototype = pn @ mbn^T  [N x N], dense row-major store.
// Block = 128 threads (4 waves) -> 64x64 output tile.
// ---------------------------------------------------------------------------
__global__ __launch_bounds__(128) void gemm_proto(
    const float* __restrict__ P, const float* __restrict__ B,
    float* __restrict__ out, int N) {
  const int lane = threadIdx.x & 31;
  const int wave = threadIdx.x >> 5;
  const int m0 = blockIdx.y * 64 + wave * 16;
  const int n0 = blockIdx.x * 64;
  const int mrow = m0 + (lane & 15);
  v8f c0 = {}, c1 = {}, c2 = {}, c3 = {};
  wmma_strip16x64(P, B, mrow, n0, lane, c0, c1, c2, c3);
  const int mbase = m0 + ((lane >> 4) << 3);  // lanes 16-31 hold M+8
  const int nl = lane & 15;
#pragma unroll
  for (int v = 0; v < 8; ++v) {
    const size_t rb = (size_t)(mbase + v) * N;
    out[rb + n0 +      nl] = c0[v];
    out[rb + n0 + 16 + nl] = c1[v];
    out[rb + n0 + 32 + nl] = c2[v];
    out[rb + n0 + 48 + nl] = c3[v];
  }
}

// ---------------------------------------------------------------------------
// Kernel 4: sim_point. G = Q @ Q^T over [2N x 2N], scattered into [N,N,2,2]:
// flat = (row>>1)*4N + (col>>1)*4 + (row&1)*2 + (col&1).
// ---------------------------------------------------------------------------
__global__ __launch_bounds__(128) void gemm_point(
    const float* __restrict__ Q, float* __restrict__ out, int N) {
  const int lane = threadIdx.x & 31;
  const int wave = threadIdx.x >> 5;
  const int m0 = blockIdx.y * 64 + wave * 16;
  const int n0 = blockIdx.x * 64;
  const int mrow = m0 + (lane & 15);
  v8f c0 = {}, c1 = {}, c2 = {}, c3 = {};
  wmma_strip16x64(Q, Q, mrow, n0, lane, c0, c1, c2, c3);
  const int mbase = m0 + ((lane >> 4) << 3);
  const int nl = lane & 15;
  const size_t rowpitch = (size_t)N * 4;
#pragma unroll
  for (int v = 0; v < 8; ++v) {
    const int row = mbase + v;
    const size_t rb = (size_t)(row >> 1) * rowpitch + (size_t)(row & 1) * 2;
    const int col0 = n0 + nl;
    out[rb + (size_t)((col0      ) >> 1) * 4 + ((col0      ) & 1)] = c0[v];
    out[rb + (size_t)((col0 + 16) >> 1) * 4 + ((col0 + 16) & 1)] = c1[v];
    out[rb + (size_t)((col0 + 32) >> 1) * 4 + ((col0 + 32) & 1)] = c2[v];
    out[rb + (size_t)((col0 + 48) >> 1) * 4 + ((col0 + 48) & 1)] = c3[v];
  }
}

// ---------------------------------------------------------------------------
extern "C" void kernel_launch(void* const* d_in, const int* in_sizes, int n_in,
                              void* d_out, int out_size, void* d_ws, size_t ws_size,
                              hipStream_t stream) {
  const float* features    = (const float*)d_in[0];   // [N, 16, 256]
  const float* prototypes  = (const float*)d_in[1];   // [N, 256]
  const float* memory_bank = (const float*)d_in[2];   // [N, 256]
  const int N = in_sizes[1] / C_DIM;                  // 4096

  // workspace: pn [N,256] | mbn [N,256] | Q [2N,256]  = 16 MB
  float* pn  = (float*)d_ws;
  float* mbn = pn + (size_t)N * C_DIM;
  float* Q   = mbn + (size_t)N * C_DIM;

  float* sim_proto = (float*)d_out;                         // [N, N]
  float* sim_point = sim_proto + (size_t)N * N;             // [N, N, 2, 2]

  normalize_rows<<<2 * N, 256, 0, stream>>>(prototypes, memory_bank, pn, mbn, N);
  mine_hardest<<<N, 256, 0, stream>>>(features, Q);

  dim3 g1(N / 64, N / 64);
  gemm_proto<<<g1, 128, 0, stream>>>(pn, mbn, sim_proto, N);

  dim3 g2((2 * N) / 64, (2 * N) / 64);
  gemm_point<<<g2, 128, 0, stream>>>(Q, sim_point, N);
}